// GAT_57887569215518
// MI455X (gfx1250) — compile-verified
//
#include <hip/hip_runtime.h>
#include <hip/hip_bf16.h>
#include <math.h>

// GAT layer for MI455X (gfx1250, wave32).
// Pipeline:
//   K1: h = X (N x 128) @ Wflat^T (128 x 128) via V_WMMA_F32_16X16X4_F32
//   K2: e_src[n,k] = <h[n,k,:], a_src[k,:]>, e_dst likewise (shuffle reduce)
//   K3: per-node leaky-relu logits -> softmax over 16 neighbors -> weighted
//       gather-sum of h[nbr] (L2-resident, 51MB << 192MB L2) -> ELU -> out.

typedef __attribute__((ext_vector_type(2))) float v2f;
typedef __attribute__((ext_vector_type(8))) float v8f;

#define DEG    16
#define KHEADS 4
#define F_IN   128
#define F_OUT  32
#define FTOT   128   // KHEADS * F_OUT

// ---------------------------------------------------------------------------
// Kernel 1: h = X @ Wflat^T.
// Wflat[j][i] = W[j/32][j%32][i]  (j = k*32+f), contiguous in i -> row of W.
// Block = 256 threads = 8 waves; wave w owns output cols [16w, 16w+16).
// Each wave keeps its 128x16 B-slice in 32 v2f registers and sweeps 4
// row-tiles of 16 rows, doing 32 f32 WMMAs per tile (K = 128 = 32 * 4).
// WMMA f32 16x16x4 operand layout (ISA 7.12.2):
//   A: lane&15 = M, K = 2*(lane>>4) + {0,1} in the two VGPRs
//   B: lane&15 = N, same K striping
//   D: VGPR r -> M = r + 8*(lane>>4), N = lane&15
// ---------------------------------------------------------------------------
__global__ __launch_bounds__(256) void gat_gemm_h(const float* __restrict__ X,
                                                  const float* __restrict__ W,
                                                  float* __restrict__ h,
                                                  int numRowTiles) {
  const int lane = threadIdx.x & 31;
  const int wv   = threadIdx.x >> 5;   // 0..7 : which 16-col tile
  const int half = lane >> 4;          // 0/1  : K-half selector
  const int l16  = lane & 15;
  const int col  = wv * 16 + l16;      // output column j = k*32 + f

  // Stage B slice in registers: b[kc] holds Wflat[col][4*kc + 2*half + {0,1}]
  v2f b[32];
  const float* wcol = W + (size_t)col * F_IN;
  #pragma unroll
  for (int kc = 0; kc < 32; ++kc)
    b[kc] = *(const v2f*)(wcol + kc * 4 + half * 2);

  #pragma unroll 1
  for (int t = 0; t < 4; ++t) {
    const int tile = blockIdx.x * 4 + t;      // wave-uniform guard, EXEC stays ~0
    if (tile >= numRowTiles) break;
    const int row = tile * 16 + l16;
    const float* xrow = X + (size_t)row * F_IN;
    v8f c = {};
    #pragma unroll
    for (int kc = 0; kc < 32; ++kc) {
      v2f a = *(const v2f*)(xrow + kc * 4 + half * 2);
      c = __builtin_amdgcn_wmma_f32_16x16x4_f32(false, a, false, b[kc],
                                                (short)0, c, false, false);
    }
    float* hout = h + (size_t)(tile * 16 + 8 * half) * FTOT + col;
    #pragma unroll
    for (int r = 0; r < 8; ++r)
      hout[(size_t)r * FTOT] = c[r];
  }
}

// ---------------------------------------------------------------------------
// Kernel 2: attention logit halves. One wave per node; lane owns 4 features
// (float4 load of h), head = lane>>3; 8-lane xor-shuffle reduction.
// a layout: a[k][0][0:32] = a_src, a[k][0][32:64] = a_dst.
// ---------------------------------------------------------------------------
__global__ __launch_bounds__(256) void gat_edge_scores(const float* __restrict__ h,
                                                       const float* __restrict__ a,
                                                       float* __restrict__ e_src,
                                                       float* __restrict__ e_dst,
                                                       int n) {
  const int lane = threadIdx.x & 31;
  const int node = blockIdx.x * 8 + (threadIdx.x >> 5);
  if (node >= n) return;
  const float4 hv = *(const float4*)(h + (size_t)node * FTOT + lane * 4);
  const int kh = lane >> 3;
  const int f0 = (lane & 7) * 4;
  const float* ak = a + kh * 2 * F_OUT;
  float s = hv.x * ak[f0]         + hv.y * ak[f0 + 1]
          + hv.z * ak[f0 + 2]     + hv.w * ak[f0 + 3];
  float t = hv.x * ak[F_OUT + f0]     + hv.y * ak[F_OUT + f0 + 1]
          + hv.z * ak[F_OUT + f0 + 2] + hv.w * ak[F_OUT + f0 + 3];
  #pragma unroll
  for (int off = 1; off < 8; off <<= 1) {
    s += __shfl_xor(s, off, 32);
    t += __shfl_xor(t, off, 32);
  }
  if ((lane & 7) == 0) {
    e_src[node * KHEADS + kh] = s;
    e_dst[node * KHEADS + kh] = t;
  }
}

// ---------------------------------------------------------------------------
// Kernel 3: softmax over neighbors + weighted gather-sum + ELU.
// One wave per node; lane owns 4 contiguous output features (one head each),
// so the DEG=16 softmax is entirely lane-local (no shuffles, no LDS).
// Gathers hit L2 (h is 51MB, L2 is 192MB).
// ---------------------------------------------------------------------------
__global__ __launch_bounds__(256) void gat_attn(const float* __restrict__ h,
                                                const int* __restrict__ nbr,
                                                const float* __restrict__ e_src,
                                                const float* __restrict__ e_dst,
                                                float* __restrict__ out,
                                                int n) {
  const int lane = threadIdx.x & 31;
  const int node = blockIdx.x * 8 + (threadIdx.x >> 5);
  if (node >= n) return;
  const int kh = lane >> 3;

  int idx[DEG];
  const int4* nb = (const int4*)(nbr + (size_t)node * DEG);
  #pragma unroll
  for (int q = 0; q < DEG / 4; ++q) {
    const int4 v = nb[q];
    idx[q * 4 + 0] = v.x; idx[q * 4 + 1] = v.y;
    idx[q * 4 + 2] = v.z; idx[q * 4 + 3] = v.w;
  }

  const float es = e_src[node * KHEADS + kh];
  float e[DEG];
  float m = -3.402823466e38f;
  #pragma unroll
  for (int d = 0; d < DEG; ++d) {
    float v = es + e_dst[idx[d] * KHEADS + kh];
    v = v > 0.f ? v : 0.01f * v;             // leaky_relu(., 0.01)
    e[d] = v;
    m = fmaxf(m, v);
  }
  float ssum = 0.f;
  #pragma unroll
  for (int d = 0; d < DEG; ++d) { e[d] = __expf(e[d] - m); ssum += e[d]; }
  const float inv = 1.f / ssum;

  float4 acc = {0.f, 0.f, 0.f, 0.f};
  const int foff = lane * 4;
  #pragma unroll
  for (int d = 0; d < DEG; ++d) {
    const float4 hv = *(const float4*)(h + (size_t)idx[d] * FTOT + foff);
    const float al = e[d] * inv;
    acc.x += al * hv.x; acc.y += al * hv.y;
    acc.z += al * hv.z; acc.w += al * hv.w;
  }
  // ELU (alpha = 1)
  acc.x = acc.x > 0.f ? acc.x : __expf(acc.x) - 1.f;
  acc.y = acc.y > 0.f ? acc.y : __expf(acc.y) - 1.f;
  acc.z = acc.z > 0.f ? acc.z : __expf(acc.z) - 1.f;
  acc.w = acc.w > 0.f ? acc.w : __expf(acc.w) - 1.f;
  *(float4*)(out + (size_t)node * FTOT + foff) = acc;
}

// ---------------------------------------------------------------------------
// Launcher. Inputs: X (N*128 f32), W (4*32*128 f32), a (4*64 f32),
// nbr (N*16 i32). Output: N*128 f32. Workspace: h (N*128 f32) +
// e_src/e_dst (N*4 f32 each) = ~54.4 MB.
// ---------------------------------------------------------------------------
extern "C" void kernel_launch(void* const* d_in, const int* in_sizes, int n_in,
                              void* d_out, int out_size, void* d_ws, size_t ws_size,
                              hipStream_t stream) {
  const float* X   = (const float*)d_in[0];
  const float* W   = (const float*)d_in[1];
  const float* a   = (const float*)d_in[2];
  const int*   nbr = (const int*)d_in[3];
  float* out = (float*)d_out;

  const int n = in_sizes[0] / F_IN;   // N = 100000 (divisible by 16)

  char* ws = (char*)d_ws;
  float* h     = (float*)ws;
  float* e_src = (float*)(ws + (size_t)n * FTOT * sizeof(float));
  float* e_dst = e_src + (size_t)n * KHEADS;

  const int rowTiles = (n + 15) / 16;
  const int gridGemm = (rowTiles + 3) / 4;          // 4 row-tiles per block
  gat_gemm_h<<<gridGemm, 256, 0, stream>>>(X, W, h, rowTiles);

  const int gridNode = (n + 7) / 8;                 // 8 waves = 8 nodes / block
  gat_edge_scores<<<gridNode, 256, 0, stream>>>(h, a, e_src, e_dst, n);
  gat_attn<<<gridNode, 256, 0, stream>>>(h, nbr, e_src, e_dst, out, n);
}